// PFELayer_9844065042483
// MI455X (gfx1250) — compile-verified
//
#include <hip/hip_runtime.h>
#include <stdint.h>

// Problem constants from the reference.
#define GRID_H 640
#define GRID_W 640
#define NBINS  (GRID_H * GRID_W)   // 409600 bins, 4 channels each

typedef float v4f __attribute__((ext_vector_type(4)));

// ---------------------------------------------------------------------------
// Native CDNA5 float max atomic (no return): GLOBAL_ATOMIC_MAX_NUM_F32.
// IEEE maximumNumber RMW performed at device scope (L2-resident bin region,
// 6.55 MB << 192 MB global L2). Tracked by STOREcnt; S_ENDPGM's implicit
// wait-idle covers completion. "memory" clobber orders it vs surrounding code.
// ---------------------------------------------------------------------------
__device__ __forceinline__ void atomic_max_f32_dev(float* p, float v) {
    asm volatile("global_atomic_max_num_f32 %0, %1, off scope:SCOPE_DEV"
                 :
                 : "v"(p), "v"(v)
                 : "memory");
}

// Zero-init the segment-max output region (must run every launch: harness
// poisons once, atomics below only ever raise values above 0). Regular RT
// stores: these lines are about to become the hot atomic working set in L2.
__global__ void pfe_zero_bins(v4f* __restrict__ seg4, int n4) {
    int i = blockIdx.x * blockDim.x + threadIdx.x;
    if (i < n4) seg4[i] = (v4f){0.f, 0.f, 0.f, 0.f};
}

// One thread per point: NT b128 load, mask, NT b128 masked store, up to 4 f32
// max-num atomics into the bin grid. Values <= 0 are skipped — they cannot
// affect max(seg, 0). Non-temporal hints keep the 256 MB streaming traffic
// from evicting the 6.55 MB bin region out of the 192 MB L2.
__global__ void pfe_scatter_max(const v4f* __restrict__ pc,
                                const float* __restrict__ lim,   // [l00,l01,l10,l11]
                                const int*   __restrict__ size,  // [640,640]
                                v4f*         __restrict__ pc_masked,
                                float*       __restrict__ seg,
                                int n) {
    int i = blockIdx.x * blockDim.x + threadIdx.x;
    if (i >= n) return;

    const float l00 = lim[0], l01 = lim[1], l10 = lim[2], l11 = lim[3];
    const int   s0  = size[0];
    const float s0f = (float)size[0];
    const float s1f = (float)size[1];

    v4f p = __builtin_nontemporal_load(pc + i);   // global_load_b128 th:NT

    const bool m = (p.x > l00) && (p.x < l01) && (p.y > l10) && (p.y < l11);

    v4f o = m ? p : (v4f){0.f, 0.f, 0.f, 0.f};
    __builtin_nontemporal_store(o, pc_masked + i); // global_store_b128 th:NT

    if (m) {
        // Bit-exact replication of the reference index math:
        // truncating int32 cast of ((v - lo) / (hi - lo)) * float(size)
        int xi = (int)(((p.x - l00) / (l01 - l00)) * s0f);
        int yi = (int)(((p.y - l10) / (l11 - l10)) * s1f);
        int idx = xi * s0 + yi;
        if (idx >= 0 && idx < NBINS) {  // segment_max drops out-of-range ids
            float* b = seg + (size_t)idx * 4;
            if (p.x > 0.f) atomic_max_f32_dev(b + 0, p.x);
            if (p.y > 0.f) atomic_max_f32_dev(b + 1, p.y);
            if (p.z > 0.f) atomic_max_f32_dev(b + 2, p.z);
            if (p.w > 0.f) atomic_max_f32_dev(b + 3, p.w);
        }
    }
}

extern "C" void kernel_launch(void* const* d_in, const int* in_sizes, int n_in,
                              void* d_out, int out_size, void* d_ws, size_t ws_size,
                              hipStream_t stream) {
    const v4f*   pc   = (const v4f*)  d_in[0];  // (N,4) f32
    const float* lim  = (const float*)d_in[1];  // (2,2) f32
    const int*   size = (const int*)  d_in[2];  // (2,)  i32

    const int n = in_sizes[0] / 4;              // N points

    float* out       = (float*)d_out;
    v4f*   pc_masked = (v4f*)out;               // first N*4 floats
    float* seg       = out + (size_t)n * 4;     // then NBINS*4 floats (16B aligned)

    // 1) Zero the bin region (NBINS v4f = 409600 stores).
    {
        const int n4 = NBINS;
        const int threads = 256;                // 8 wave32 per block
        const int blocks  = (n4 + threads - 1) / threads;
        pfe_zero_bins<<<blocks, threads, 0, stream>>>((v4f*)seg, n4);
    }

    // 2) Masked copy + scatter f32-max into bins.
    {
        const int threads = 256;
        const int blocks  = (n + threads - 1) / threads;
        pfe_scatter_max<<<blocks, threads, 0, stream>>>(pc, lim, size,
                                                        pc_masked, seg, n);
    }
}